// _EdgeWeightsGraphConvLayer_63651415327483
// MI455X (gfx1250) — compile-verified
//
#include <hip/hip_runtime.h>
#include <hip/hip_bf16.h>

#define N_ELEC 19
#define EDGES_PER_GRAPH 361
#define C 64   // C_IN == C_OUT == 64

typedef float v2f __attribute__((ext_vector_type(2)));
typedef float v8f __attribute__((ext_vector_type(8)));

// ---------------------------------------------------------------------------
// Kernel 1: zero the aggregation buffer (d_out doubles as agg storage)
// ---------------------------------------------------------------------------
__global__ void _egc_zero_f4(float4* __restrict__ p, int n4) {
    int i = blockIdx.x * blockDim.x + threadIdx.x;
    int stride = gridDim.x * blockDim.x;
    float4 z = make_float4(0.f, 0.f, 0.f, 0.f);
    for (; i < n4; i += stride) p[i] = z;
}

// ---------------------------------------------------------------------------
// Kernel 2: gather-scale-scatter.  32 lanes per edge, 2 channels per lane.
// agg[dst] += ew[e % 361] * x[src]  via L2 float atomics (data set is
// L2-resident on MI455X: x = 40MB, agg = 40MB << 192MB L2).
// ---------------------------------------------------------------------------
__global__ void _egc_edge_scatter(const float* __restrict__ x,
                                  const int*   __restrict__ src,
                                  const int*   __restrict__ dst,
                                  const float* __restrict__ ew,
                                  float*       __restrict__ agg,
                                  int E) {
    int gid  = blockIdx.x * blockDim.x + threadIdx.x;
    int e    = gid >> 5;              // edge index (32 lanes per edge)
    if (e >= E) return;
    int lane = gid & 31;
    int c    = lane << 1;             // 2 channels per lane -> coalesced 8B

    int   s = src[e];
    int   d = dst[e];
    float w = ew[e % EDGES_PER_GRAPH];

    float2 xv = *reinterpret_cast<const float2*>(x + (size_t)s * C + c);
    float* ap = agg + (size_t)d * C + c;
    unsafeAtomicAdd(ap,     w * xv.x);   // global_atomic_add_f32
    unsafeAtomicAdd(ap + 1, w * xv.y);
}

// ---------------------------------------------------------------------------
// Kernel 3: out = agg @ W_rel^T + b_rel + x @ W_root^T, in place on d_out.
// V_WMMA_F32_16X16X4_F32, full fp32.  One block = 16 rows; 4 waves each own a
// 16-wide output-channel group.  agg/x tiles staged in LDS (pad 66 to kill
// bank conflicts on the strided A-matrix reads), sync, then compute+store —
// makes the in-place update of d_out race-free within the block.
// ---------------------------------------------------------------------------
#if __has_builtin(__builtin_amdgcn_wmma_f32_16x16x4_f32)
#define EGC_HAVE_WMMA_F32 1
#endif

__global__ __launch_bounds__(128)
void _egc_graphconv_gemm(float*       __restrict__ out,    // agg in, result out
                         const float* __restrict__ x,
                         const float* __restrict__ Wrel,   // [64][64] row-major
                         const float* __restrict__ brel,   // [64]
                         const float* __restrict__ Wroot)  // [64][64] row-major
{
    __shared__ float sAgg[16 * 66];
    __shared__ float sX  [16 * 66];

    const int    tid  = threadIdx.x;
    const size_t row0 = (size_t)blockIdx.x * 16;

    // cooperative stage of the 16x64 agg and x tiles (fully coalesced)
    #pragma unroll
    for (int i = 0; i < 8; ++i) {
        int idx = tid + i * 128;
        int r   = idx >> 6;
        int col = idx & 63;
        sAgg[r * 66 + col] = out[(row0 + r) * C + col];
        sX  [r * 66 + col] = x  [(row0 + r) * C + col];
    }
    __syncthreads();

    const int wave = tid >> 5;
    const int lane = tid & 31;
    const int nl   = lane & 15;            // A row m / C-D column-in-tile
    const int koff = (lane >> 4) << 1;     // lanes 16-31 carry K+2,K+3
    const int n    = wave * 16 + nl;       // global output channel

    const float bias = brel[n];
    v8f acc = { bias, bias, bias, bias, bias, bias, bias, bias };

    // B[k][n] = W[n][k] (W^T), so lane n reads row n of W: float2 at even k.
    const float2* wrel2  = reinterpret_cast<const float2*>(Wrel  + (size_t)n * C);
    const float2* wroot2 = reinterpret_cast<const float2*>(Wroot + (size_t)n * C);

#ifdef EGC_HAVE_WMMA_F32
    #pragma unroll
    for (int kb = 0; kb < C; kb += 4) {
        const int k0 = kb + koff;

        v2f aA, aX, bR, bT;
        aA.x = sAgg[nl * 66 + k0];  aA.y = sAgg[nl * 66 + k0 + 1];
        aX.x = sX  [nl * 66 + k0];  aX.y = sX  [nl * 66 + k0 + 1];
        float2 wr = wrel2 [k0 >> 1];
        float2 wt = wroot2[k0 >> 1];
        bR.x = wr.x; bR.y = wr.y;
        bT.x = wt.x; bT.y = wt.y;

        // D = A*B + C  (args: neg_a, A, neg_b, B, c_mod, C, reuse_a, reuse_b)
        acc = __builtin_amdgcn_wmma_f32_16x16x4_f32(
                  false, aA, false, bR, (short)0, acc, false, false);
        acc = __builtin_amdgcn_wmma_f32_16x16x4_f32(
                  false, aX, false, bT, (short)0, acc, false, false);
    }

    // C/D layout: VGPR r -> M = r + (lane<16 ? 0 : 8), N = lane%16
    const int mb = (lane >> 4) * 8;
    #pragma unroll
    for (int r = 0; r < 8; ++r)
        out[(row0 + mb + r) * C + n] = acc[r];
#else
    // scalar fallback (not expected to be used: builtin probe-confirmed)
    #pragma unroll
    for (int i = 0; i < 8; ++i) {
        int idx = tid * 8 + i;
        int m   = idx >> 6;
        int nn  = idx & 63;
        float s = brel[nn];
        for (int k = 0; k < C; ++k)
            s += sAgg[m * 66 + k] * Wrel[nn * C + k]
               + sX  [m * 66 + k] * Wroot[nn * C + k];
        out[(row0 + m) * C + nn] = s;
    }
#endif
}

// ---------------------------------------------------------------------------
// launch
// ---------------------------------------------------------------------------
extern "C" void kernel_launch(void* const* d_in, const int* in_sizes, int n_in,
                              void* d_out, int out_size, void* d_ws, size_t ws_size,
                              hipStream_t stream) {
    const float* x     = (const float*)d_in[0];
    const int*   ei    = (const int*)  d_in[1];   // [2, E] (src row, dst row)
    const float* ew    = (const float*)d_in[2];   // [361]
    const float* Wrel  = (const float*)d_in[3];   // [64,64]
    const float* brel  = (const float*)d_in[4];   // [64]
    const float* Wroot = (const float*)d_in[5];   // [64,64]
    float*       out   = (float*)d_out;

    const int nNodes = in_sizes[0] / C;           // 155648 (divisible by 16)
    const int E      = in_sizes[1] / 2;           // 2957312

    // 1) agg = 0 (d_out is the agg buffer; must be re-zeroed every call)
    const int n4 = (nNodes * C) / 4;
    _egc_zero_f4<<<4096, 256, 0, stream>>>((float4*)out, n4);

    // 2) scatter-add messages: 32 lanes / edge
    const long long threads = (long long)E * 32;
    const int blocks = (int)((threads + 255) / 256);
    _egc_edge_scatter<<<blocks, 256, 0, stream>>>(x, ei, ei + (size_t)E, ew, out, E);

    // 3) fused dual-GEMM + bias via f32 WMMA, in place on d_out
    _egc_graphconv_gemm<<<nNodes / 16, 128, 0, stream>>>(out, x, Wrel, brel, Wroot);
}